// RandomSample_64707977282334
// MI455X (gfx1250) — compile-verified
//
#include <hip/hip_runtime.h>

// Problem constants (from reference): B=4, N=16384, DIMS=3, F_DIMS=64
#define BATCH 4
#define NPTS  16384
#define HALF  8192
#define FD    64
#define SHPAD (HALF + 16)   // LDS panel + 1 tile of padding for branch-free prefetch

typedef __attribute__((ext_vector_type(2))) float v2f;
typedef __attribute__((ext_vector_type(8))) float v8f;
typedef unsigned long long u64;
typedef unsigned int u32;

// Pointer types for the async global->LDS builtin (per hipcc's diagnostic the
// param is "int __vector(4) __device__ *", i.e. vector_size(16) int in AS(1);
// the LDS side is the AS(3) counterpart).
typedef int vec4i __attribute__((vector_size(16)));
typedef __attribute__((address_space(1))) vec4i gvec4i;
typedef __attribute__((address_space(3))) vec4i lvec4i;

// ---------------------------------------------------------------------------
// Kernel 1: gather valid/invalid points.
//  - valid_pc    -> d_out[0 .. B*HALF*3)
//  - valid_feats -> d_out[B*HALF*3 .. +B*HALF*64)
//  - B4[t] = (-2x, -2y, -2z, x^2+y^2+z^2)  for valid point t   (GEMM B side)
//  - A4[t] = ( x,   y,   z,  1)            for invalid point t (GEMM A side)
// With C = q^2 fed into the WMMA accumulator, D = q^2 - 2 q.r + r^2 = d^2 >= 0,
// so d^2 bit patterns are monotone as u32 and pack into sortable u64 keys.
// ---------------------------------------------------------------------------
__global__ __launch_bounds__(256)
void gather_prep(const float* __restrict__ pc,
                 const float* __restrict__ feats,
                 const int*   __restrict__ rnds,
                 float*       __restrict__ out,
                 float4*      __restrict__ A4,
                 float4*      __restrict__ B4)
{
    int t = blockIdx.x * blockDim.x + threadIdx.x;
    if (t >= BATCH * HALF) return;
    int b = t / HALF;
    int j = t % HALF;
    const int* rb = rnds + b * NPTS;

    // ---- valid point: copy pc, feats; build B-side vector ----
    int vi = rb[j];
    const float* p = pc + (size_t)(b * NPTS + vi) * 3;
    float x = p[0], y = p[1], z = p[2];
    float* opc = out + (size_t)(b * HALF + j) * 3;
    opc[0] = x; opc[1] = y; opc[2] = z;
    B4[t] = make_float4(-2.0f * x, -2.0f * y, -2.0f * z, x * x + y * y + z * z);

    // feats row: 64 floats = 16 float4 (rows are 256B aligned)
    const float4* fs = (const float4*)(feats + (size_t)(b * NPTS + vi) * FD);
    float4* fdst = (float4*)(out + (size_t)(BATCH * HALF * 3)
                                 + (size_t)(b * HALF + j) * FD);
#pragma unroll
    for (int k = 0; k < FD / 4; ++k) fdst[k] = fs[k];

    // ---- invalid (query) point: build A-side homogeneous vector ----
    int ui = rb[HALF + j];
    const float* q = pc + (size_t)(b * NPTS + ui) * 3;
    A4[t] = make_float4(q[0], q[1], q[2], 1.0f);
}

// ---------------------------------------------------------------------------
// Kernel 2: per query, index of the 2nd nearest valid point.
//
// Block = 8 waves, each wave owns 16 query rows; all waves sweep the same
// 8192-column B panel, staged once into LDS via the gfx1250 async global->LDS
// path, then read with conflict-free ds_load_b64, software-pipelined one tile
// ahead so the ds wait is covered by the previous tile's WMMA + key updates.
//
// Wave layout (wave32):
//   A fragment (16x4 f32): lanes 0-15 hold row M=lane, K0/K1; lanes 16-31
//     hold row M=lane-16, K2/K3  -> float2 load at [2*row + hi].
//   B fragment (4x16 f32, mirrored): lane (L&15) = column, halves = K pairs.
//   C fragment = q^2 per row (v8f), built once with 8 shuffles.
//   D (16x16 f32): lane L holds column N=(L&15); element i is row i+8*(L>=16).
//
// Top-2 as sortable u64 keys (smax(d2_bits,0) << 32 | col) maintained with
// native v_min_u64 / v_max_u64 — fully branchless.
// ---------------------------------------------------------------------------
__global__ __launch_bounds__(256)
void knn2_wmma(const float4* __restrict__ A4,
               const float4* __restrict__ B4,
               int*          __restrict__ nidx)
{
    extern __shared__ float4 Bs[];                       // SHPAD float4

    const int lane = threadIdx.x & 31;
    const int wave = threadIdx.x >> 5;
    const int blocksPerBatch = HALF / 128;               // 64
    const int b     = blockIdx.x / blocksPerBatch;
    const int qblk  = blockIdx.x % blocksPerBatch;
    const int qbase = qblk * 128 + wave * 16;            // 16 rows per wave
    const int hi    = (lane >= 16) ? 1 : 0;
    const int lj    = lane & 15;

    // ---- stage the whole B panel for this batch into LDS ----
    const float4* Bg = B4 + (size_t)b * HALF;
#if __has_builtin(__builtin_amdgcn_global_load_async_to_lds_b128)
    for (int i = threadIdx.x; i < HALF; i += 256) {
        __builtin_amdgcn_global_load_async_to_lds_b128(
            (gvec4i*)(void*)(Bg + i), (lvec4i*)(void*)(Bs + i), 0, 0);
    }
#if __has_builtin(__builtin_amdgcn_s_wait_asynccnt)
    __builtin_amdgcn_s_wait_asynccnt(0);
#else
    asm volatile("s_wait_asynccnt 0" ::: "memory");
#endif
#else
    for (int i = threadIdx.x; i < HALF; i += 256) Bs[i] = Bg[i];
#endif
    __syncthreads();

    // ---- loop-invariant A fragment: float2 load of this lane's K half ----
    const float2* Ap = (const float2*)(A4 + (size_t)b * HALF + qbase);
    float2 af = Ap[2 * lj + hi];                         // (x,y) or (z,1)
    v2f a; a.x = af.x; a.y = af.y;

    // ---- loop-invariant C fragment: q^2 of rows (hi?8:0)+0..7 ----
    float4 av = A4[(size_t)b * HALF + qbase + lj];
    float q2 = av.x * av.x + av.y * av.y + av.z * av.z;
    v8f cq;
#pragma unroll
    for (int i = 0; i < 8; ++i)
        cq[i] = __shfl(q2, i + hi * 8, 32);

    u64 best[8], sec[8];
#pragma unroll
    for (int i = 0; i < 8; ++i) { best[i] = ~0ull; sec[i] = ~0ull; }

    const float2* Bs2 = (const float2*)Bs;

    // Software pipeline: bv holds tile t, bn prefetches tile t+1 (pad-backed).
    float2 bv = Bs2[2 * lj + hi];

#pragma unroll 4
    for (int c0 = 0; c0 < HALF; c0 += 16) {
        const int col = c0 + lj;
        float2 bn = Bs2[2 * (col + 16) + hi];            // ds_load_b64, next tile

        v2f bf; bf.x = bv.x; bf.y = bv.y;

        // D = A(16x4) x B(4x16) + q^2  -> true squared distances (>= 0)
        v8f d = __builtin_amdgcn_wmma_f32_16x16x4_f32(
            /*neg_a=*/false, a, /*neg_b=*/false, bf,
            /*c_mod=*/(short)0, cq, /*reuse_a=*/false, /*reuse_b=*/false);

        const u64 colb = (u32)col;
#pragma unroll
        for (int i = 0; i < 8; ++i) {
            // Integer clamp: any negative rounding artifact -> +0.0 bits, and
            // non-negative f32 bit patterns are u32-monotone.  One v_max_i32.
            int db  = __builtin_elementwise_max(__float_as_int(d[i]), 0);
            u64 key = ((u64)(u32)db << 32) | colb;
            u64 b0  = best[i];
            u64 nb  = __builtin_elementwise_min(key, b0);
            u64 mx  = __builtin_elementwise_max(key, b0);
            best[i] = nb;
            sec[i]  = __builtin_elementwise_min(sec[i], mx);
        }
        bv = bn;
    }

    // ---- butterfly merge of top-2 keys across the 16 lanes per row set ----
#pragma unroll
    for (int m = 1; m <= 8; m <<= 1) {
#pragma unroll
        for (int i = 0; i < 8; ++i) {
            u64 ob = __shfl_xor(best[i], m, 32);
            u64 os = __shfl_xor(sec[i],  m, 32);
            u64 nb = __builtin_elementwise_min(best[i], ob);
            u64 mx = __builtin_elementwise_max(best[i], ob);
            sec[i] = __builtin_elementwise_min(
                         __builtin_elementwise_min(sec[i], os), mx);
            best[i] = nb;
        }
    }

    // Lane 0 writes rows 0..7, lane 16 writes rows 8..15.
    if (lj == 0) {
        size_t base = (size_t)b * HALF + qbase + (hi ? 8 : 0);
#pragma unroll
        for (int i = 0; i < 8; ++i)
            nidx[base + i] = (int)(u32)(sec[i] & 0xFFFFFFFFull);
    }
}

// ---------------------------------------------------------------------------
// d_out layout (flat, reference return order):
//   [0)        valid_pc    B*HALF*3  = 98304 floats
//   [98304)    valid_feats B*HALF*64 = 2097152 floats
//   [2195456)  n_idx       B*HALF    = 32768 int32 (bit-stored)
// d_ws: A4 (B*HALF float4) then B4 (B*HALF float4) = 1 MiB total.
// ---------------------------------------------------------------------------
extern "C" void kernel_launch(void* const* d_in, const int* in_sizes, int n_in,
                              void* d_out, int out_size, void* d_ws, size_t ws_size,
                              hipStream_t stream) {
    const float* pc    = (const float*)d_in[0];
    const float* feats = (const float*)d_in[1];
    const int*   rnds  = (const int*)d_in[2];
    float* out = (float*)d_out;

    float4* A4 = (float4*)d_ws;
    float4* B4 = A4 + (size_t)BATCH * HALF;
    int* nidx = (int*)(out + (size_t)BATCH * HALF * 3 + (size_t)BATCH * HALF * FD);

    gather_prep<<<(BATCH * HALF + 255) / 256, 256, 0, stream>>>(
        pc, feats, rnds, out, A4, B4);

    knn2_wmma<<<BATCH * (HALF / 128), 256, SHPAD * sizeof(float4), stream>>>(
        A4, B4, nidx);
}